// SparseGraphConv_59399397704084
// MI455X (gfx1250) — compile-verified
//
#include <hip/hip_runtime.h>
#include <hip/hip_bf16.h>

typedef __attribute__((ext_vector_type(2))) float v2f;
typedef __attribute__((ext_vector_type(8))) float v8f;

#define D 128  // D_IN == D_OUT == 128

// ---------------------------------------------------------------------------
// Stage 1: src_out = src @ W using V_WMMA_F32_16X16X4_F32 (fp32 matrix pipe).
// Block = 256 threads = 8 waves. Each block handles a 16-row tile of src;
// wave w computes the 16x16 output tile at columns [16w, 16w+16).
// K loop: 128 / 4 = 32 WMMA steps.
//
// Fragment layouts (ISA 7.12.2, 32-bit):
//   A 16x4 : lanes 0-15 -> {K=0,K=1}, lanes 16-31 -> {K=2,K=3}, M = lane&15
//   B 4x16 : vgpr0: lanes0-15=row k0, lanes16-31=row k0+1; vgpr1: rows k0+2,k0+3
//   C/D    : vgpr i, lane<16 -> (M=i, N=lane); lane>=16 -> (M=i+8, N=lane-16)
// ---------------------------------------------------------------------------
__global__ __launch_bounds__(256) void sgc_gemm_wmma(
    const float* __restrict__ src, const float* __restrict__ W,
    float* __restrict__ out) {
  const int lane = threadIdx.x & 31;
  const int wave = threadIdx.x >> 5;   // 0..7 : N-tile index
  const int half = lane >> 4;          // 0 or 1 (lane group)
  const int l    = lane & 15;

  const int m0 = blockIdx.x * 16;
  const int n0 = wave * 16;

  const float* arow = src + (size_t)(m0 + l) * D;  // row for A fragment
  const int    col  = n0 + l;                      // column for B fragment

  v8f c = {};
#pragma unroll 4
  for (int k0 = 0; k0 < D; k0 += 4) {
    v2f a, b;
    // A: this lane holds (M = m0+l, K = k0 + 2*half + {0,1})
    a.x = arow[k0 + 2 * half + 0];
    a.y = arow[k0 + 2 * half + 1];
    // B: vgpr0 = row (k0+half), vgpr1 = row (k0+half+2), column = col
    b.x = W[(size_t)(k0 + half)     * D + col];
    b.y = W[(size_t)(k0 + half + 2) * D + col];
    c = __builtin_amdgcn_wmma_f32_16x16x4_f32(
        /*neg_a=*/false, a, /*neg_b=*/false, b,
        /*c_mod=*/(short)0, c, /*reuse_a=*/false, /*reuse_b=*/false);
  }

  // Store D-matrix: vgpr i -> row m0 + i + 8*half, column n0 + l
  float* obase = out + (size_t)(m0 + 8 * half) * D + n0 + l;
#pragma unroll
  for (int i = 0; i < 8; ++i) {
    obase[(size_t)i * D] = c[i];
  }
}

// ---------------------------------------------------------------------------
// Stage 2: msg[edge_dst] += edge_weight * src_out[edge_src]
// One wave per edge; each lane handles 4 consecutive dims (float4 gather,
// 4 native fp32 global atomic adds). Gathers of src_out mostly hit the
// 192 MB L2 (src_out is 51.2 MB).
// ---------------------------------------------------------------------------
__global__ __launch_bounds__(256) void sgc_edge_scatter(
    const float* __restrict__ src_out, const float* __restrict__ edge_weight,
    const int* __restrict__ edge_src, const int* __restrict__ edge_dst,
    float* __restrict__ msg, int n_edges) {
  const int e = blockIdx.x * 8 + (threadIdx.x >> 5);
  if (e >= n_edges) return;
  const int lane = threadIdx.x & 31;

  const float w = edge_weight[e];
  const int   s = edge_src[e];
  const int   d = edge_dst[e];

  const float4 v = ((const float4*)(src_out + (size_t)s * D))[lane];
  float* drow = msg + (size_t)d * D + lane * 4;
  unsafeAtomicAdd(drow + 0, w * v.x);
  unsafeAtomicAdd(drow + 1, w * v.y);
  unsafeAtomicAdd(drow + 2, w * v.z);
  unsafeAtomicAdd(drow + 3, w * v.w);
}

// ---------------------------------------------------------------------------
// Stage 3: out = msg / max(||msg||_2, eps), row-wise.
// One wave per row; float4 per lane; wave32 shfl_xor reduction.
// ---------------------------------------------------------------------------
__global__ __launch_bounds__(256) void sgc_normalize(
    const float* __restrict__ msg, float* __restrict__ out, int n_rows) {
  const int row = blockIdx.x * 8 + (threadIdx.x >> 5);
  if (row >= n_rows) return;
  const int lane = threadIdx.x & 31;

  const float4 v = ((const float4*)(msg + (size_t)row * D))[lane];
  float ss = v.x * v.x + v.y * v.y + v.z * v.z + v.w * v.w;
#pragma unroll
  for (int off = 16; off > 0; off >>= 1) ss += __shfl_xor(ss, off, 32);

  const float inv = 1.0f / fmaxf(sqrtf(ss), 1e-12f);
  float4 o;
  o.x = v.x * inv; o.y = v.y * inv; o.z = v.z * inv; o.w = v.w * inv;
  ((float4*)(out + (size_t)row * D))[lane] = o;
}

// ---------------------------------------------------------------------------
// Inputs (setup_inputs order): src[100000*128] f32, dst (unused), W[128*128]
// f32, edge_weight[600000] f32, edge_src[600000] i32, edge_dst[600000] i32.
// Output: normalized msg [N_DST*128] f32.
// Workspace: [src_out: N_SRC*128 f32][msg: N_DST*128 f32] = 102.4 MB.
// ---------------------------------------------------------------------------
extern "C" void kernel_launch(void* const* d_in, const int* in_sizes, int n_in,
                              void* d_out, int out_size, void* d_ws, size_t ws_size,
                              hipStream_t stream) {
  const float* src = (const float*)d_in[0];
  const float* W   = (const float*)d_in[2];
  const float* ew  = (const float*)d_in[3];
  const int*   es  = (const int*)d_in[4];
  const int*   ed  = (const int*)d_in[5];

  const int n_src   = in_sizes[0] / D;   // 100000
  const int n_dst   = out_size / D;      // 100000
  const int n_edges = in_sizes[3];       // 600000

  float* src_out = (float*)d_ws;
  float* msg     = src_out + (size_t)n_src * D;

  // Zero the segment-sum accumulator (capture-safe async memset).
  hipMemsetAsync(msg, 0, (size_t)n_dst * D * sizeof(float), stream);

  // Stage 1: dense GEMM via fp32 WMMA. 100000/16 = 6250 blocks.
  sgc_gemm_wmma<<<(n_src + 15) / 16, 256, 0, stream>>>(src, W, src_out);

  // Stage 2: weighted gather/scatter-add over edges (wave per edge).
  sgc_edge_scatter<<<(n_edges + 7) / 8, 256, 0, stream>>>(
      src_out, ew, es, ed, msg, n_edges);

  // Stage 3: row-wise L2 normalize into d_out (wave per row).
  sgc_normalize<<<(n_dst + 7) / 8, 256, 0, stream>>>(msg, (float*)d_out, n_dst);
}